// DKVMN_86028194938888
// MI455X (gfx1250) — compile-verified
//
#include <hip/hip_runtime.h>
#include <hip/hip_bf16.h>
#include <math.h>

typedef float v2f __attribute__((ext_vector_type(2)));
typedef float v8f __attribute__((ext_vector_type(8)));

#define Bsz 64
#define Tsz 100
#define Dk  128
#define Dv  128
#define Nn  128
#define NUMQ 1000
#define ROWS (Bsz*Tsz)   /* 6400 */

enum { ACT_NONE = 0, ACT_SIGMOID = 1, ACT_TANH = 2 };

// ---------------------------------------------------------------------------
// Phase 0: embedding gather.  Kg[t*B+b][:] = k_emb[q[b,t]], Vg = v_emb[q+NUMQ*r]
// ---------------------------------------------------------------------------
__global__ void gather_kv(const int* __restrict__ q, const int* __restrict__ r,
                          const float* __restrict__ k_emb, const float* __restrict__ v_emb,
                          float* __restrict__ Kg, float* __restrict__ Vg) {
  int row = blockIdx.x;              // row = t*B + b
  int t = row / Bsz, b = row - t * Bsz;
  int qi = q[b * Tsz + t];
  int vi = qi + NUMQ * r[b * Tsz + t];
  int d = threadIdx.x;
  Kg[(size_t)row * Dk + d] = k_emb[(size_t)qi * Dk + d];
  Vg[(size_t)row * Dv + d] = v_emb[(size_t)vi * Dv + d];
}

// ---------------------------------------------------------------------------
// Phase 1: f32 GEMM via V_WMMA_F32_16X16X4_F32.
// C[M x 128] = act(A[M x 128] @ B[128 x 128] + bias).  One wave per 16x16 tile;
// block = 256 threads = 8 waves = full N strip; grid.x = M/16.
// ---------------------------------------------------------------------------
template <int ACT>
__global__ __launch_bounds__(256) void gemm_wmma_f32(
    const float* __restrict__ Am, const float* __restrict__ Bm,
    const float* __restrict__ bias, float* __restrict__ Cm, int K, int N) {
  int lane = threadIdx.x & 31;
  int wave = threadIdx.x >> 5;       // 0..7 -> N tile
  int row0 = blockIdx.x * 16;
  int col0 = wave * 16;
  int mr   = lane & 15;              // A row / C column lane
  int kh   = lane >> 4;              // K half-select (0: K=0,1  1: K=2,3)

  v8f acc = {};
  const float* ap = Am + (size_t)(row0 + mr) * K + 2 * kh;
  const float* bp = Bm + (size_t)(2 * kh) * N + col0 + mr;
  for (int k0 = 0; k0 < K; k0 += 4) {
    v2f a = *(const v2f*)ap;                 // A[row][k0+2*kh+{0,1}]
    v2f b;
    b.x = bp[0];                             // B[k0+2*kh+0][col]
    b.y = bp[N];                             // B[k0+2*kh+1][col]
    ap += 4;
    bp += (size_t)4 * N;
    acc = __builtin_amdgcn_wmma_f32_16x16x4_f32(
        /*neg_a=*/false, a, /*neg_b=*/false, b,
        /*c_mod=*/(short)0, acc, /*reuse_a=*/false, /*reuse_b=*/false);
  }
  int c = col0 + mr;
  float bv = bias ? bias[c] : 0.0f;
#pragma unroll
  for (int v = 0; v < 8; ++v) {
    float x = acc[v] + bv;                   // C row = row0 + v + 8*kh
    if (ACT == ACT_SIGMOID) x = 1.0f / (1.0f + __expf(-x));
    if (ACT == ACT_TANH)    x = tanhf(x);
    Cm[(size_t)(row0 + v + 8 * kh) * N + c] = x;
  }
}

// ---------------------------------------------------------------------------
// Row softmax over N=128 (in place).
// ---------------------------------------------------------------------------
__global__ void softmax_rows(float* __restrict__ S) {
  __shared__ float red[128];
  int row = blockIdx.x, tid = threadIdx.x;
  float x = S[(size_t)row * Nn + tid];
  red[tid] = x; __syncthreads();
  for (int s = 64; s > 0; s >>= 1) {
    if (tid < s) red[tid] = fmaxf(red[tid], red[tid + s]);
    __syncthreads();
  }
  float m = red[0]; __syncthreads();
  float e = __expf(x - m);
  red[tid] = e; __syncthreads();
  for (int s = 64; s > 0; s >>= 1) {
    if (tid < s) red[tid] += red[tid + s];
    __syncthreads();
  }
  S[(size_t)row * Nn + tid] = e / red[0];
}

// ---------------------------------------------------------------------------
// Phase 2: sequential scan. One workgroup per batch; Mv state in LDS (64 KB).
// Per step: rt = wt . Mv ; pt = sigmoid(tanh(rt@fW1 + G2) @ pW + pb);
// Mv = Mv*(1 - w*e) + w*a ; stream Mv to d_out (coalesced 64 KB store).
// ---------------------------------------------------------------------------
__global__ __launch_bounds__(256) void dkvmn_scan(
    const float* __restrict__ Mv0, const float* __restrict__ Wt,
    const float* __restrict__ Et,  const float* __restrict__ At,
    const float* __restrict__ G2,  const float* __restrict__ fW1,
    const float* __restrict__ pW,  const float* __restrict__ pb,
    float* __restrict__ p_out, float* __restrict__ mv_out) {
  extern __shared__ float sm[];
  float* Mv  = sm;               // 16384 floats
  float* w_s = sm + Nn * Dv;     // 128
  float* e_s = w_s + 128;        // 128
  float* a_s = e_s + 128;        // 128
  float* g_s = a_s + 128;        // 128
  float* r_s = g_s + 128;        // 128
  float* red = r_s + 128;        // 128

  int b = blockIdx.x, tid = threadIdx.x;
  for (int i = tid; i < Nn * Dv; i += blockDim.x) Mv[i] = Mv0[i];
  __syncthreads();

  for (int t = 0; t < Tsz; ++t) {
    int row = t * Bsz + b;
    if (tid < 128) {
      w_s[tid] = Wt[(size_t)row * Nn + tid];
      e_s[tid] = Et[(size_t)row * Dv + tid];
      a_s[tid] = At[(size_t)row * Dv + tid];
      g_s[tid] = G2[(size_t)row * Dk + tid];
    }
    __syncthreads();

    if (tid < 128) {                       // rt[d] = sum_n w[n] * Mv[n][d]
      float acc = 0.0f;
      for (int n = 0; n < Nn; ++n) acc += w_s[n] * Mv[n * Dv + tid];
      r_s[tid] = acc;
    }
    __syncthreads();                       // rt ready; all Mv reads done

    // Memory update + coalesced streaming store of the new state.
    float* outp = mv_out + (size_t)row * (Nn * Dv);
    for (int i = tid; i < Nn * Dv; i += blockDim.x) {
      int n = i >> 7, d = i & 127;
      float w  = w_s[n];
      float nv = Mv[i] * (1.0f - w * e_s[d]) + w * a_s[d];
      Mv[i] = nv;
      outp[i] = nv;
    }

    // ft = tanh(rt @ fW1 + G2); contribution to p reduction.
    if (tid < 128) {
      float acc = g_s[tid];
      for (int k = 0; k < Dk; ++k) acc += r_s[k] * fW1[k * Dk + tid];
      red[tid] = tanhf(acc) * pW[tid];
    }
    __syncthreads();
    for (int s = 64; s > 0; s >>= 1) {
      if (tid < s) red[tid] += red[tid + s];
      __syncthreads();
    }
    if (tid == 0) p_out[row] = 1.0f / (1.0f + __expf(-(red[0] + pb[0])));
    __syncthreads();
  }
}

// ---------------------------------------------------------------------------
extern "C" void kernel_launch(void* const* d_in, const int* in_sizes, int n_in,
                              void* d_out, int out_size, void* d_ws, size_t ws_size,
                              hipStream_t stream) {
  const int*   q     = (const int*)d_in[0];
  const int*   r     = (const int*)d_in[1];
  const float* k_emb = (const float*)d_in[2];
  const float* v_emb = (const float*)d_in[3];
  const float* Mk    = (const float*)d_in[4];
  const float* Mv0   = (const float*)d_in[5];
  const float* f_W   = (const float*)d_in[6];
  const float* f_b   = (const float*)d_in[7];
  const float* p_W   = (const float*)d_in[8];
  const float* p_b   = (const float*)d_in[9];
  const float* e_W   = (const float*)d_in[10];
  const float* e_b   = (const float*)d_in[11];
  const float* a_W   = (const float*)d_in[12];
  const float* a_b   = (const float*)d_in[13];

  float* ws = (float*)d_ws;
  float* Kg = ws;                              // 6400x128
  float* Vg = Kg + (size_t)ROWS * Dk;          // 6400x128
  float* S  = Vg + (size_t)ROWS * Dv;          // 6400x128 (scores -> softmax w)
  float* E  = S  + (size_t)ROWS * Nn;          // 6400x128
  float* A  = E  + (size_t)ROWS * Dv;          // 6400x128
  float* G2 = A  + (size_t)ROWS * Dv;          // 6400x128

  float* p_out  = (float*)d_out;               // 6400
  float* mv_out = p_out + ROWS;                // 6400*128*128

  gather_kv<<<ROWS, 128, 0, stream>>>(q, r, k_emb, v_emb, Kg, Vg);

  gemm_wmma_f32<ACT_NONE><<<ROWS / 16, 256, 0, stream>>>(Kg, Mk, nullptr, S, Dk, Nn);
  softmax_rows<<<ROWS, 128, 0, stream>>>(S);
  gemm_wmma_f32<ACT_SIGMOID><<<ROWS / 16, 256, 0, stream>>>(Vg, e_W, e_b, E, Dv, Dv);
  gemm_wmma_f32<ACT_TANH><<<ROWS / 16, 256, 0, stream>>>(Vg, a_W, a_b, A, Dv, Dv);
  // G2 = Kg @ f_W[128:,:] + f_b   (the kt half of the concat matmul)
  gemm_wmma_f32<ACT_NONE><<<ROWS / 16, 256, 0, stream>>>(Kg, f_W + (size_t)Dk * Dk, f_b, G2, Dk, Dk);

  size_t smem = (size_t)(Nn * Dv + 6 * 128) * sizeof(float);   // ~68.6 KB
  dkvmn_scan<<<Bsz, 256, smem, stream>>>(Mv0, S, E, A, G2, f_W, p_W, p_b, p_out, mv_out);
}